// BiMpmMatching_84731114815853
// MI455X (gfx1250) — compile-verified
//
#include <hip/hip_runtime.h>
#include <hip/hip_bf16.h>
#include <math.h>

// Only check for the gfx1250 WMMA builtin during the DEVICE compilation pass;
// the host (x86) pass never declares amdgcn builtins.
#if defined(__HIP_DEVICE_COMPILE__)
#if !__has_builtin(__builtin_amdgcn_wmma_f32_16x16x4_f32)
#error "gfx1250 f32 WMMA builtin __builtin_amdgcn_wmma_f32_16x16x4_f32 not available on device"
#endif
#endif

// Problem constants (match setup_inputs)
constexpr int B = 8, L = 256, H = 128, P = 20;
constexpr int OD = 252;                 // output last dim (126 per side)
constexpr float EPSV = 1e-8f;
constexpr float NEGV = -1e30f;

typedef float v2f __attribute__((ext_vector_type(2)));
typedef float v8f __attribute__((ext_vector_type(8)));

__device__ inline v8f wmma_f32(v2f a, v2f b, v8f c) {
  // D = A(16x4) x B(4x16) + C, all f32, wave32.
  return __builtin_amdgcn_wmma_f32_16x16x4_f32(false, a, false, b, (short)0, c, false, false);
}

__device__ inline float wave_sum(float x) {
  #pragma unroll
  for (int m = 16; m >= 1; m >>= 1) x += __shfl_xor(x, m, 32);
  return x;
}
__device__ inline float wave_max(float x) {
  #pragma unroll
  for (int m = 16; m >= 1; m >>= 1) x = fmaxf(x, __shfl_xor(x, m, 32));
  return x;
}

// ---------------- prologue kernels ----------------

__global__ void k_mask(const float* __restrict__ c1, const float* __restrict__ c2,
                       const int* __restrict__ m1, const int* __restrict__ m2,
                       float* __restrict__ c1m, float* __restrict__ c2m) {
  int t = blockIdx.x * blockDim.x + threadIdx.x;     // over B*L*H
  if (t >= B * L * H) return;
  int bl = t / H;
  c1m[t] = (m1[bl] > 0) ? c1[t] : 0.f;
  c2m[t] = (m2[bl] > 0) ? c2[t] : 0.f;
}

__global__ void k_w2(const float* __restrict__ wmp, float* __restrict__ w2) {
  int t = blockIdx.x * blockDim.x + threadIdx.x;
  if (t < P * H) { float w = wmp[t]; w2[t] = w * w; }
}

// Row norms + per-perspective weighted norms. One wave per (b,row), grid.y = side.
__global__ void k_norms(const float* __restrict__ c1m, const float* __restrict__ c2m,
                        const float* __restrict__ w2,
                        float* __restrict__ n1, float* __restrict__ n2,
                        float* __restrict__ n1p, float* __restrict__ n2p) {
  int side = blockIdx.y;
  int wav = threadIdx.x >> 5, lane = threadIdx.x & 31;
  int bl = blockIdx.x * 8 + wav;
  const float* c = (side == 0 ? c1m : c2m) + (size_t)bl * H;
  int h0 = lane * 4;
  float v[4];
  #pragma unroll
  for (int q = 0; q < 4; q++) v[q] = c[h0 + q];
  float s = v[0]*v[0] + v[1]*v[1] + v[2]*v[2] + v[3]*v[3];
  s = wave_sum(s);
  if (lane == 0) (side == 0 ? n1 : n2)[bl] = sqrtf(s);
  int b = bl / L, i = bl % L;
  float* np = (side == 0 ? n1p : n2p);
  for (int p = 0; p < P; p++) {
    const float* wr = w2 + p * H + h0;
    float sp = wr[0]*v[0]*v[0] + wr[1]*v[1]*v[1] + wr[2]*v[2]*v[2] + wr[3]*v[3]*v[3];
    sp = wave_sum(sp);
    if (lane == 0) np[(b * P + p) * L + i] = sqrtf(sp);
  }
}

// Mask counts + first/last valid index per (b, side). idx layout: [s1 | e1 | s2 | e2], B each.
__global__ void k_meta(const int* __restrict__ m1, const int* __restrict__ m2,
                       float* __restrict__ cnt1, float* __restrict__ cnt2,
                       int* __restrict__ idx) {
  int t = threadIdx.x;
  if (t >= 2 * B) return;
  int side = t / B, b = t % B;
  const int* m = (side == 0 ? m1 : m2) + b * L;
  int c = 0, s = 0, e = L - 1; bool found = false;
  for (int j = 0; j < L; j++)
    if (m[j] > 0) { c++; if (!found) { s = j; found = true; } e = j; }
  if (!found) { s = 0; e = L - 1; }   // matches argmax-of-all-false behavior
  if (side == 0) { cnt1[b] = (float)c; idx[b] = s; idx[B + b] = e; }
  else           { cnt2[b] = (float)c; idx[2*B + b] = s; idx[3*B + b] = e; }
}

// ---------------- cosine matrix (WMMA f32) ----------------
// cs[b,i,j] = (c1m[b,i,:] . c2m[b,j,:]) / (max(|c1|,eps)*max(|c2|,eps))
__global__ void k_cs(const float* __restrict__ c1m, const float* __restrict__ c2m,
                     const float* __restrict__ n1, const float* __restrict__ n2,
                     float* __restrict__ cs) {
  int b = blockIdx.y;
  int wav = threadIdx.x >> 5, lane = threadIdx.x & 31;
  int tile = blockIdx.x * 8 + wav;            // 0..255 -> 16x16 tiles of 256x256
  int ti = tile >> 4, tj = tile & 15;
  int r = lane & 15, hf = lane >> 4;
  const float* Ar = c1m + (size_t)(b * L + ti * 16 + r) * H;
  const float* Br = c2m + (size_t)(b * L + tj * 16 + r) * H;
  v8f acc = {};
  for (int kk = 0; kk < H; kk += 4) {
    int k = kk + 2 * hf;
    v2f a = { Ar[k], Ar[k + 1] };
    v2f bb = { Br[k], Br[k + 1] };
    acc = wmma_f32(a, bb, acc);
  }
  int j = tj * 16 + r;
  float nj = fmaxf(n2[b * L + j], EPSV);
  #pragma unroll
  for (int v = 0; v < 8; v++) {
    int i = ti * 16 + hf * 8 + v;
    float ni = fmaxf(n1[b * L + i], EPSV);
    cs[((size_t)(b * L + i)) * L + j] = acc[v] / (ni * nj);
  }
}

// masked max/mean over cs rows (side0) and columns (side1) -> out cols {0,1}/{126,127}
__global__ void k_csred(const float* __restrict__ cs,
                        const int* __restrict__ m1, const int* __restrict__ m2,
                        const float* __restrict__ cnt1, const float* __restrict__ cnt2,
                        float* __restrict__ out) {
  int side = blockIdx.y;
  int wav = threadIdx.x >> 5, lane = threadIdx.x & 31;
  int bl = blockIdx.x * 8 + wav;
  int b = bl / L, i = bl % L;
  const int* mo = (side == 0 ? m2 : m1) + b * L;
  float mx = NEGV, sm = 0.f;
  for (int j0 = 0; j0 < L; j0 += 32) {
    int j = j0 + lane;
    float v = (side == 0) ? cs[((size_t)(b * L + i)) * L + j]
                          : cs[((size_t)(b * L + j)) * L + i];
    bool ok = mo[j] > 0;
    mx = fmaxf(mx, ok ? v : NEGV);
    sm += ok ? v : 0.f;
  }
  mx = wave_max(mx); sm = wave_sum(sm);
  if (lane == 0) {
    float cnt = (side == 0 ? cnt2 : cnt1)[b];
    float* o = out + (size_t)bl * OD + (side == 0 ? 0 : 126);
    o[0] = mx;
    o[1] = sm / fmaxf(cnt, EPSV);
  }
}

// ---------------- attentive matmuls (WMMA f32) ----------------
// side0: attraw2[b,i,h] = sum_j cs[b,i,j]*c2m[b,j,h]
// side1: attraw1[b,j,h] = sum_i cs[b,i,j]*c1m[b,i,h]
__global__ void k_attmm(const float* __restrict__ cs,
                        const float* __restrict__ c1m, const float* __restrict__ c2m,
                        float* __restrict__ attraw2, float* __restrict__ attraw1) {
  int side = blockIdx.z;
  int b = blockIdx.y;
  int wav = threadIdx.x >> 5, lane = threadIdx.x & 31;
  int tile = blockIdx.x * 8 + wav;            // 0..127: 16 row-tiles x 8 col-tiles
  int ti = tile >> 3, tn = tile & 7;
  int r = lane & 15, hf = lane >> 4;
  const float* csb = cs + (size_t)b * L * L;
  const float* Cm = (side == 0 ? c2m : c1m) + (size_t)b * L * H;
  int ar = ti * 16 + r;                       // A row (output row) for this lane
  int nc = tn * 16 + r;                       // B col (h) for this lane
  v8f acc = {};
  for (int kk = 0; kk < L; kk += 4) {
    int k = kk + 2 * hf;
    v2f a, bb;
    if (side == 0) a = (v2f){ csb[(size_t)ar * L + k], csb[(size_t)ar * L + k + 1] };
    else           a = (v2f){ csb[(size_t)k * L + ar], csb[(size_t)(k + 1) * L + ar] };
    bb = (v2f){ Cm[(size_t)k * H + nc], Cm[(size_t)(k + 1) * H + nc] };
    acc = wmma_f32(a, bb, acc);
  }
  float* O = (side == 0 ? attraw2 : attraw1) + (size_t)b * L * H;
  #pragma unroll
  for (int v = 0; v < 8; v++) {
    int row = ti * 16 + hf * 8 + v;
    O[(size_t)row * H + tn * 16 + r] = acc[v];
  }
}

// masked softmax over H (mask broadcast along H), times mask
__global__ void k_softmax(const float* __restrict__ attraw2, const float* __restrict__ attraw1,
                          const int* __restrict__ m1, const int* __restrict__ m2,
                          float* __restrict__ attmean2, float* __restrict__ attmean1) {
  int side = blockIdx.y;
  int wav = threadIdx.x >> 5, lane = threadIdx.x & 31;
  int bl = blockIdx.x * 8 + wav;
  const float* X = (side == 0 ? attraw2 : attraw1) + (size_t)bl * H;
  float* O = (side == 0 ? attmean2 : attmean1) + (size_t)bl * H;
  bool valid = ((side == 0 ? m1 : m2)[bl]) > 0;
  int h0 = lane * 4;
  float v[4], e[4];
  #pragma unroll
  for (int q = 0; q < 4; q++) v[q] = X[h0 + q];
  float mx = fmaxf(fmaxf(v[0], v[1]), fmaxf(v[2], v[3]));
  mx = wave_max(mx);
  float s = 0.f;
  #pragma unroll
  for (int q = 0; q < 4; q++) { e[q] = expf(v[q] - mx); s += e[q]; }
  s = wave_sum(s);
  #pragma unroll
  for (int q = 0; q < 4; q++) O[h0 + q] = valid ? e[q] / s : 0.f;
}

// max-attentive vectors: masked max over the opposite sequence of c*cs
__global__ void k_attmax(const float* __restrict__ cs,
                         const float* __restrict__ c1m, const float* __restrict__ c2m,
                         const int* __restrict__ m1, const int* __restrict__ m2,
                         float* __restrict__ attmax2, float* __restrict__ attmax1) {
  int side = blockIdx.y;
  int bl = blockIdx.x;
  int b = bl / L, i = bl % L;
  int h = threadIdx.x;                        // 128 threads
  const float* csb = cs + (size_t)b * L * L;
  float acc = NEGV;
  if (side == 0) {
    const float* C = c2m + (size_t)b * L * H;
    const int* m = m2 + b * L;
    for (int j = 0; j < L; j++) {
      float val = C[(size_t)j * H + h] * csb[(size_t)i * L + j];
      acc = fmaxf(acc, m[j] > 0 ? val : NEGV);
    }
    attmax2[(size_t)bl * H + h] = acc;
  } else {
    const float* C = c1m + (size_t)b * L * H;
    const int* m = m1 + b * L;
    for (int j = 0; j < L; j++) {             // j iterates side-1 positions
      float val = C[(size_t)j * H + h] * csb[(size_t)j * L + i];
      acc = fmaxf(acc, m[j] > 0 ? val : NEGV);
    }
    attmax1[(size_t)bl * H + h] = acc;
  }
}

// ---------------- maxpool match (WMMA f32 + fused reductions) ----------------
// One block per (b,p): full 256x256 perspective-weighted cosine tile, reduced to
// row max/mean (mv1 cols 44..83) and col max/mean (mv2 cols 170..209) in-block.
__global__ void k_maxpool(const float* __restrict__ c1m, const float* __restrict__ c2m,
                          const float* __restrict__ w2,
                          const float* __restrict__ n1p, const float* __restrict__ n2p,
                          const int* __restrict__ m1, const int* __restrict__ m2,
                          const float* __restrict__ cnt1, const float* __restrict__ cnt2,
                          float* __restrict__ out) {
  __shared__ float rowmax[L], rowsum[L];
  __shared__ float colmax[8][L], colsum[8][L];
  __shared__ float w2s[H];
  int b = blockIdx.x, p = blockIdx.y;
  int tid = threadIdx.x;
  int wav = tid >> 5, lane = tid & 31;
  int r = lane & 15, hf = lane >> 4;

  if (tid < H) w2s[tid] = w2[p * H + tid];
  rowmax[tid] = NEGV; rowsum[tid] = 0.f;
  #pragma unroll
  for (int w = 0; w < 8; w++) { colmax[w][tid] = NEGV; colsum[w][tid] = 0.f; }
  __syncthreads();

  const float* A0 = c1m + (size_t)b * L * H;
  const float* B0 = c2m + (size_t)b * L * H;
  const int* mm1 = m1 + b * L;
  const int* mm2 = m2 + b * L;
  const float* np1 = n1p + (size_t)(b * P + p) * L;
  const float* np2 = n2p + (size_t)(b * P + p) * L;

  for (int tt = 0; tt < 2; tt++) {
    int ti = wav * 2 + tt;                    // wave-private tile rows
    const float* Ar = A0 + (size_t)(ti * 16 + r) * H;
    for (int tj = 0; tj < 16; tj++) {
      const float* Br = B0 + (size_t)(tj * 16 + r) * H;
      v8f acc = {};
      for (int kk = 0; kk < H; kk += 4) {
        int k = kk + 2 * hf;
        v2f a = { w2s[k] * Ar[k], w2s[k + 1] * Ar[k + 1] };
        v2f bb = { Br[k], Br[k + 1] };
        acc = wmma_f32(a, bb, acc);
      }
      int j = tj * 16 + r;
      bool okj = mm2[j] > 0;
      float denj = np2[j];
      float cm = NEGV, csum = 0.f;
      float vals[8];
      #pragma unroll
      for (int v = 0; v < 8; v++) {
        int i = ti * 16 + hf * 8 + v;
        float den = fmaxf(np1[i] * denj, EPSV);   // product-eps clamp (ref _mpm_pairwise)
        float val = acc[v] / den;
        vals[v] = val;
        bool oki = mm1[i] > 0;
        cm = fmaxf(cm, oki ? val : NEGV);
        csum += oki ? val : 0.f;
      }
      // column reduction: combine the two row-halves
      cm = fmaxf(cm, __shfl_xor(cm, 16, 32));
      csum += __shfl_xor(csum, 16, 32);
      if (hf == 0) {
        colmax[wav][j] = fmaxf(colmax[wav][j], cm);
        colsum[wav][j] += csum;
      }
      // row reductions over the 16 columns of this tile
      #pragma unroll
      for (int v = 0; v < 8; v++) {
        float rv = okj ? vals[v] : NEGV;
        float sv = okj ? vals[v] : 0.f;
        #pragma unroll
        for (int m = 8; m >= 1; m >>= 1) {
          rv = fmaxf(rv, __shfl_xor(rv, m, 32));
          sv += __shfl_xor(sv, m, 32);
        }
        if (r == 0) {
          int i = ti * 16 + hf * 8 + v;
          rowmax[i] = fmaxf(rowmax[i], rv);
          rowsum[i] += sv;
        }
      }
    }
  }
  __syncthreads();
  float c2c = fmaxf(cnt2[b], EPSV);
  float c1c = fmaxf(cnt1[b], EPSV);
  float* o = out + (size_t)(b * L + tid) * OD;
  o[44 + p] = rowmax[tid];
  o[64 + p] = rowsum[tid] / c2c;
  float gm = NEGV, gs = 0.f;
  #pragma unroll
  for (int w = 0; w < 8; w++) { gm = fmaxf(gm, colmax[w][tid]); gs += colsum[w][tid]; }
  o[126 + 44 + p] = gm;
  o[126 + 64 + p] = gs / c1c;
}

// ---------------- multi-perspective match (8 instances), one wave per (b,l) ----------------
__global__ void k_mpm(const float* __restrict__ c1m, const float* __restrict__ c2m,
                      const float* __restrict__ attmean2, const float* __restrict__ attmean1,
                      const float* __restrict__ attmax2, const float* __restrict__ attmax1,
                      const float* __restrict__ wf, const float* __restrict__ wb,
                      const float* __restrict__ wa, const float* __restrict__ wma,
                      const int* __restrict__ idx, float* __restrict__ out) {
  int wav = threadIdx.x >> 5, lane = threadIdx.x & 31;
  int bl = blockIdx.x * 8 + wav;
  int b = bl / L;
  int h0 = lane * 4;
  float v1a[4], v1b[4];
  #pragma unroll
  for (int q = 0; q < 4; q++) {
    v1a[q] = c1m[(size_t)bl * H + h0 + q];
    v1b[q] = c2m[(size_t)bl * H + h0 + q];
  }
  int s1 = idx[b], e1 = idx[B + b], s2 = idx[2 * B + b], e2 = idx[3 * B + b];

  const float* v2s[8] = {
    c2m + (size_t)(b * L + e2) * H,      // mv1: full fwd vs last-of-2
    c2m + (size_t)(b * L + s2) * H,      // mv1: full bwd vs first-of-2
    attmean2 + (size_t)bl * H,           // mv1: attentive
    attmax2 + (size_t)bl * H,            // mv1: max-attentive
    c1m + (size_t)(b * L + e1) * H,      // mv2: full fwd vs last-of-1
    c1m + (size_t)(b * L + s1) * H,      // mv2: full bwd vs first-of-1
    attmean1 + (size_t)bl * H,
    attmax1 + (size_t)bl * H
  };
  const float* wsel[8] = { wf, wb, wa, wma, wf, wb, wa, wma };
  const int obase[8] = { 2, 23, 84, 105, 126 + 2, 126 + 23, 126 + 84, 126 + 105 };

  float* o = out + (size_t)bl * OD;
  for (int c = 0; c < 8; c++) {
    float u[4], t[4];
    #pragma unroll
    for (int q = 0; q < 4; q++) {
      u[q] = (c < 4) ? v1a[q] : v1b[q];
      t[q] = v2s[c][h0 + q];
    }
    float d = 0.f, na = 0.f, nb = 0.f;
    #pragma unroll
    for (int q = 0; q < 4; q++) { d += u[q] * t[q]; na += u[q] * u[q]; nb += t[q] * t[q]; }
    d = wave_sum(d); na = wave_sum(na); nb = wave_sum(nb);
    if (lane == 0)
      o[obase[c]] = d / (fmaxf(sqrtf(na), EPSV) * fmaxf(sqrtf(nb), EPSV));
    for (int p = 0; p < P; p++) {
      const float* wr = wsel[c] + p * H + h0;
      float dp = 0.f, nap = 0.f, nbp = 0.f;
      #pragma unroll
      for (int q = 0; q < 4; q++) {
        float w = wr[q];
        float wu = w * u[q], wt = w * t[q];
        dp += wu * wt; nap += wu * wu; nbp += wt * wt;
      }
      dp = wave_sum(dp); nap = wave_sum(nap); nbp = wave_sum(nbp);
      if (lane == 0)
        o[obase[c] + 1 + p] = dp / (fmaxf(sqrtf(nap), EPSV) * fmaxf(sqrtf(nbp), EPSV));
    }
  }
}

// ---------------- launch ----------------

extern "C" void kernel_launch(void* const* d_in, const int* in_sizes, int n_in,
                              void* d_out, int out_size, void* d_ws, size_t ws_size,
                              hipStream_t stream) {
  const float* c1  = (const float*)d_in[0];
  const float* c2  = (const float*)d_in[1];
  const int*   m1  = (const int*)d_in[2];
  const int*   m2  = (const int*)d_in[3];
  const float* wf  = (const float*)d_in[4];
  const float* wb  = (const float*)d_in[5];
  const float* wmp = (const float*)d_in[6];
  const float* wa  = (const float*)d_in[7];
  const float* wma = (const float*)d_in[8];
  float* out = (float*)d_out;

  char* base = (char*)d_ws;
  auto alloc = [&](size_t n) -> float* {
    float* pp = (float*)base; base += n * sizeof(float); return pp;
  };
  float* c1m      = alloc((size_t)B * L * H);
  float* c2m      = alloc((size_t)B * L * H);
  float* n1       = alloc((size_t)B * L);
  float* n2       = alloc((size_t)B * L);
  float* n1p      = alloc((size_t)B * P * L);
  float* n2p      = alloc((size_t)B * P * L);
  float* w2       = alloc((size_t)P * H);
  float* cs       = alloc((size_t)B * L * L);
  float* attraw2  = alloc((size_t)B * L * H);
  float* attraw1  = alloc((size_t)B * L * H);
  float* attmean2 = alloc((size_t)B * L * H);
  float* attmean1 = alloc((size_t)B * L * H);
  float* attmax2  = alloc((size_t)B * L * H);
  float* attmax1  = alloc((size_t)B * L * H);
  float* cnt1     = alloc((size_t)B);
  float* cnt2     = alloc((size_t)B);
  int*   idx      = (int*)base;  base += 4 * B * sizeof(int);
  (void)ws_size; (void)in_sizes; (void)n_in; (void)out_size;

  k_mask   <<<(B * L * H + 255) / 256, 256, 0, stream>>>(c1, c2, m1, m2, c1m, c2m);
  k_w2     <<<(P * H + 255) / 256, 256, 0, stream>>>(wmp, w2);
  k_norms  <<<dim3(B * L / 8, 2), 256, 0, stream>>>(c1m, c2m, w2, n1, n2, n1p, n2p);
  k_meta   <<<1, 32, 0, stream>>>(m1, m2, cnt1, cnt2, idx);
  k_cs     <<<dim3(32, B), 256, 0, stream>>>(c1m, c2m, n1, n2, cs);
  k_csred  <<<dim3(B * L / 8, 2), 256, 0, stream>>>(cs, m1, m2, cnt1, cnt2, out);
  k_attmm  <<<dim3(16, B, 2), 256, 0, stream>>>(cs, c1m, c2m, attraw2, attraw1);
  k_softmax<<<dim3(B * L / 8, 2), 256, 0, stream>>>(attraw2, attraw1, m1, m2, attmean2, attmean1);
  k_attmax <<<dim3(B * L, 2), 128, 0, stream>>>(cs, c1m, c2m, m1, m2, attmax2, attmax1);
  k_maxpool<<<dim3(B, P), 256, 0, stream>>>(c1m, c2m, w2, n1p, n2p, m1, m2, cnt1, cnt2, out);
  k_mpm    <<<B * L / 8, 256, 0, stream>>>(c1m, c2m, attmean2, attmean1, attmax2, attmax1,
                                           wf, wb, wa, wma, idx, out);
}